// Embedding_40200893891072
// MI455X (gfx1250) — compile-verified
//
#include <hip/hip_runtime.h>

typedef __attribute__((ext_vector_type(2))) float v2f;
typedef __attribute__((ext_vector_type(4))) float v4f;
typedef __attribute__((ext_vector_type(8))) float v8f;

#define T_DIM 4096
#define C_DIM 64
#define D_DIM 512
#define INV_DET (1.0f / 512.0f)   // det_i = sum_d exp(0) = D = 512 exactly

// ---------------------------------------------------------------------------
// Kernel 1: one workgroup (8 waves). Waves 0-3 compute rs = W * ones via
// V_WMMA_F32_16X16X4_F32 (each wave owns a 16-row tile, 128 K-steps of K=4).
// Waves 4-7 compute cs = ones^T * W (each wave owns 8 column tiles of 16,
// 16 K-steps over the 64 rows). Then all 256 threads emit
// remap[i][j] = exp(rs[i]*cs[j]) / 512 into d_ws (128 KB).
// ---------------------------------------------------------------------------
__global__ __launch_bounds__(256) void softreorder_setup(
    const float* __restrict__ W, float* __restrict__ remap) {
  __shared__ float s_rs[C_DIM];
  __shared__ float s_cs[D_DIM];

  const int tid  = threadIdx.x;
  const int wave = tid >> 5;     // 0..7
  const int lane = tid & 31;
  const int n    = lane & 15;    // M (A) or N (B/D) position
  const int hi   = lane >> 4;    // lane group
  const int koff = hi * 2;       // 32-bit A/B layout: group 1 holds K=2,3

  if (wave < 4) {
    // rs: A = 16x4 tile of W, B = ones(4x16) -> every D column = row sum
    const int row_base = wave * 16;
    v2f ones; ones.x = 1.0f; ones.y = 1.0f;
    v8f acc = {};
    for (int kb = 0; kb < D_DIM; kb += 4) {
      v2f a;
      a.x = W[(row_base + n) * D_DIM + kb + koff];
      a.y = W[(row_base + n) * D_DIM + kb + koff + 1];
      acc = __builtin_amdgcn_wmma_f32_16x16x4_f32(false, a, false, ones,
                                                  (short)0, acc, false, false);
    }
    // D layout: VGPR v, lane group hi -> M = v + 8*hi; N = n (all N equal).
    if (n == 0) {
#pragma unroll
      for (int v = 0; v < 8; ++v) s_rs[row_base + hi * 8 + v] = acc[v];
    }
  } else {
    // cs: A = ones(16x4), B = 4x16 tile of W -> every D row = column sum
    const int w4 = wave - 4;     // 0..3
    v2f ones; ones.x = 1.0f; ones.y = 1.0f;
    for (int ct = w4 * 8; ct < w4 * 8 + 8; ++ct) {
      const int col = ct * 16 + n;
      v8f acc = {};
      for (int kb = 0; kb < C_DIM; kb += 4) {
        v2f bm;
        bm.x = W[(kb + koff) * D_DIM + col];
        bm.y = W[(kb + koff + 1) * D_DIM + col];
        acc = __builtin_amdgcn_wmma_f32_16x16x4_f32(false, ones, false, bm,
                                                    (short)0, acc, false, false);
      }
      if (hi == 0) s_cs[col] = acc[0];   // any row of D; take M=0
    }
  }
  __syncthreads();

  for (int idx = tid; idx < C_DIM * D_DIM; idx += 256) {
    const int i = idx >> 9;            // / D_DIM
    const int j = idx & (D_DIM - 1);
    remap[idx] = expf(s_rs[i] * s_cs[j]) * INV_DET;
  }
}

// ---------------------------------------------------------------------------
// Kernel 2: store-bandwidth bound broadcast. out[t][i][j] = remap[i][j].
// Each thread loads one float4 of remap (128 KB set -> L2-resident) and
// streams it to 8 timesteps with non-temporal B128 stores (output is 512 MiB,
// never re-read: bypass cache rinse, feed HBM at line rate).
// ---------------------------------------------------------------------------
__global__ __launch_bounds__(256) void broadcast_kernel(
    const v4f* __restrict__ remap4, v4f* __restrict__ out4) {
  const int R4 = (C_DIM * D_DIM) / 4;                 // 8192 float4 per t
  const long g = (long)blockIdx.x * 256 + threadIdx.x; // [0, R4 * T/8)
  const int  r     = (int)(g & (long)(R4 - 1));
  const long tbase = (g >> 13) << 3;                  // 8 timesteps per thread
  v4f v = remap4[r];
  v4f* p = out4 + tbase * R4 + r;
#pragma unroll
  for (int k = 0; k < 8; ++k) {
    __builtin_nontemporal_store(v, p);
    p += R4;
  }
}

extern "C" void kernel_launch(void* const* d_in, const int* in_sizes, int n_in,
                              void* d_out, int out_size, void* d_ws, size_t ws_size,
                              hipStream_t stream) {
  (void)in_sizes; (void)n_in; (void)out_size; (void)ws_size;
  const float* W = (const float*)d_in[1];   // d_in[0] = x (only supplies T)
  float* remap = (float*)d_ws;              // 32768 floats = 128 KB scratch

  softreorder_setup<<<1, 256, 0, stream>>>(W, remap);

  const int total4  = T_DIM * (C_DIM * D_DIM / 4);    // 33,554,432 float4 stores
  const int threads = total4 / 8;                      // 8 stores per thread
  broadcast_kernel<<<threads / 256, 256, 0, stream>>>((const v4f*)remap,
                                                      (v4f*)d_out);
}